// MultiTaskUNetWithRegionDetection_41343355191650
// MI455X (gfx1250) — compile-verified
//
#include <hip/hip_runtime.h>
#include <hip/hip_bf16.h>

typedef __attribute__((ext_vector_type(2))) float v2f;
typedef __attribute__((ext_vector_type(8))) float v8f;

#define ENC_BSTRIDE (512 * 1024)   // [b][ch][32][32]
#define OH_ELEMS    (4 * 6 * 512 * 512)
#define SAG_ELEMS   (4 * 512 * 32 * 32)

// ---------------------------------------------------------------------------
// Kernel 1: rasterize one_hot_map (4,6,512,512). Pure bandwidth, float4 writes.
// ---------------------------------------------------------------------------
__global__ void k_rasterize(const int* __restrict__ bboxes,
                            float* __restrict__ onehot) {
    int idx = blockIdx.x * blockDim.x + threadIdx.x;   // one float4 (4 x-pixels)
    // per (b,c) image: 512*512/4 = 65536 float4s
    int x4 = idx & 127;
    int y  = (idx >> 7) & 511;
    int bc = idx >> 16;
    int c  = bc % 6;
    int b  = bc / 6;
    float4 v = make_float4(0.f, 0.f, 0.f, 0.f);
    if (c != 0) {
        const int* bb = bboxes + (b * 5 + (c - 1)) * 4;
        int xmin = min(max(bb[0], 0), 511);
        int ymin = min(max(bb[1], 0), 511);
        int xmax = max(xmin, min(bb[2], 512));
        int ymax = max(ymin, min(bb[3], 512));
        if (y >= ymin && y < ymax) {
            int xb = x4 * 4;
            v.x = (xb + 0 >= xmin && xb + 0 < xmax) ? 1.f : 0.f;
            v.y = (xb + 1 >= xmin && xb + 1 < xmax) ? 1.f : 0.f;
            v.z = (xb + 2 >= xmin && xb + 2 < xmax) ? 1.f : 0.f;
            v.w = (xb + 3 >= xmin && xb + 3 < xmax) ? 1.f : 0.f;
        }
    }
    reinterpret_cast<float4*>(onehot)[idx] = v;
}

// ---------------------------------------------------------------------------
// Kernel 2: X_dec_k (4,512,32,32). Evaluate conv3x3-s2 + relu only at the
// bilinear sample lattice. Block = (b, oy); 256 threads cover (ch, ox).
// ---------------------------------------------------------------------------
__global__ void k_xdeck(const int* __restrict__ bboxes,
                        const float* __restrict__ Wproc,   // (512,6,3,3)
                        const float* __restrict__ bproc,   // (512)
                        float* __restrict__ Xk) {
    __shared__ float sInd[2 * 32 * 2 * 45];   // [cy][ox][cx][j] 0/1 indicators
    __shared__ int   sBox[5][4];
    __shared__ int   sX0[32];
    __shared__ float sWx[32];

    const int b  = blockIdx.x >> 5;
    const int oy = blockIdx.x & 31;
    const int tid = threadIdx.x;

    if (tid < 5) {
        const int* bb = bboxes + (b * 5 + tid) * 4;
        int xmin = min(max(bb[0], 0), 511);
        int ymin = min(max(bb[1], 0), 511);
        int xmax = max(xmin, min(bb[2], 512));
        int ymax = max(ymin, min(bb[3], 512));
        sBox[tid][0] = xmin; sBox[tid][1] = ymin;
        sBox[tid][2] = xmax; sBox[tid][3] = ymax;
    }
    if (tid < 32) {
        double xs = (double)tid * 255.0 / 31.0;
        int x0 = (int)xs;                  // floor, xs >= 0
        sX0[tid] = x0;
        sWx[tid] = (float)(xs - (double)x0);
    }
    double ysd = (double)oy * 255.0 / 31.0;
    const int   y0 = (int)ysd;
    const float wy = (float)(ysd - (double)y0);
    const int   y1 = min(y0 + 1, 255);
    __syncthreads();

    // Fill indicator table: 2*32*2*45 = 5760 entries
    for (int e = tid; e < 5760; e += 256) {
        int j  = e % 45;
        int r  = e / 45;           // 0..127
        int cx = r & 1;
        int ox = (r >> 1) & 31;
        int cy = r >> 6;
        int x0 = sX0[ox];
        int xx = cx ? min(x0 + 1, 255) : x0;
        int yy = cy ? y1 : y0;
        int i  = j / 9;            // organ 0..4 (channel i+1)
        int kk = j % 9;
        int iy = 2 * yy - 1 + kk / 3;
        int ix = 2 * xx - 1 + kk % 3;
        float val = 0.f;
        if (iy >= 0 && iy < 512 && ix >= 0 && ix < 512) {
            val = (iy >= sBox[i][1] && iy < sBox[i][3] &&
                   ix >= sBox[i][0] && ix < sBox[i][2]) ? 1.f : 0.f;
        }
        sInd[e] = val;
    }
    __syncthreads();

    const int ox = tid & 31;          // one 32-thread group per channel slice
    const int o0 = tid >> 5;          // 0..7, loop o += 8
    const float wx = sWx[ox];
    const float* ind00 = &sInd[((0 * 32 + ox) * 2 + 0) * 45];
    const float* ind01 = &sInd[((0 * 32 + ox) * 2 + 1) * 45];
    const float* ind10 = &sInd[((1 * 32 + ox) * 2 + 0) * 45];
    const float* ind11 = &sInd[((1 * 32 + ox) * 2 + 1) * 45];

    for (int o = o0; o < 512; o += 8) {
        const float* wrow = Wproc + o * 54 + 9;  // skip bg channel (all-zero)
        float a00 = 0.f, a01 = 0.f, a10 = 0.f, a11 = 0.f;
        #pragma unroll
        for (int j = 0; j < 45; ++j) {
            float w = wrow[j];
            a00 = fmaf(w, ind00[j], a00);
            a01 = fmaf(w, ind01[j], a01);
            a10 = fmaf(w, ind10[j], a10);
            a11 = fmaf(w, ind11[j], a11);
        }
        float bias = bproc[o];
        a00 = fmaxf(a00 + bias, 0.f);
        a01 = fmaxf(a01 + bias, 0.f);
        a10 = fmaxf(a10 + bias, 0.f);
        a11 = fmaxf(a11 + bias, 0.f);
        float left  = a00 * (1.f - wy) + a10 * wy;
        float right = a01 * (1.f - wy) + a11 * wy;
        float val   = left * (1.f - wx) + right * wx;
        Xk[b * ENC_BSTRIDE + o * 1024 + oy * 32 + ox] = val;
    }
}

// ---------------------------------------------------------------------------
// Kernel 3: SAG. logits via V_WMMA_F32_16X16X4_F32 (full-fp32 matrix op),
// softmax(6) + 1x1 expand + sigmoid gate scalar. 32 blocks x 256 threads;
// each wave owns a 16-position tile (M=16, N=16 padded, K=512).
// ---------------------------------------------------------------------------
__global__ void k_sag(const float* __restrict__ enc,
                      const float* __restrict__ Walpha,   // (6,512)
                      const float* __restrict__ Wexpand,  // (512,6)
                      const float* __restrict__ Xk,
                      float* __restrict__ out) {
    __shared__ float sWa[512 * 16];      // B matrix, [k][n], n>=6 zero-padded
    __shared__ float sWe[512 * 6];
    __shared__ float sLog[128 * 17];     // logits -> alpha, [row=pos][class]

    const int tid = threadIdx.x;
    const int b       = blockIdx.x >> 3;
    const int posBase = (blockIdx.x & 7) << 7;   // 128 positions per block

    for (int e = tid; e < 512 * 16; e += 256) {
        int k = e >> 4, n = e & 15;
        sWa[e] = (n < 6) ? Walpha[n * 512 + k] : 0.f;
    }
    for (int e = tid; e < 512 * 6; e += 256) sWe[e] = Wexpand[e];
    __syncthreads();

    const int lane = tid & 31;
    const int wave = tid >> 5;
    const int m    = lane & 15;       // M index / N index
    const int half = lane >> 4;       // K sub-block select
    const int pos  = posBase + wave * 16 + m;
    const float* encB = enc + b * ENC_BSTRIDE + pos;
    const float* xkB  = Xk  + b * ENC_BSTRIDE + pos;

    v8f acc = {};
    for (int k0 = 0; k0 < 512; k0 += 4) {
        int kA = k0 + half * 2;
        v2f a, bv;
        a.x  = encB[kA * 1024]       * xkB[kA * 1024];
        a.y  = encB[(kA + 1) * 1024] * xkB[(kA + 1) * 1024];
        bv.x = sWa[kA * 16 + m];
        bv.y = sWa[(kA + 1) * 16 + m];
        acc = __builtin_amdgcn_wmma_f32_16x16x4_f32(
            /*neg_a=*/false, a, /*neg_b=*/false, bv,
            /*c_mod=*/(short)0, acc, /*reuse_a=*/false, /*reuse_b=*/false);
    }
    // D layout: VGPR v, lanes 0-15 -> M=v, lanes 16-31 -> M=v+8; N = lane%16
    #pragma unroll
    for (int v = 0; v < 8; ++v) {
        int row = v + half * 8;
        sLog[(wave * 16 + row) * 17 + m] = acc[v];
    }
    __syncthreads();

    if (tid < 128) {   // softmax over 6 classes, one row per thread
        float* L = &sLog[tid * 17];
        float mx = L[0];
        #pragma unroll
        for (int c = 1; c < 6; ++c) mx = fmaxf(mx, L[c]);
        float ex[6], s = 0.f;
        #pragma unroll
        for (int c = 0; c < 6; ++c) { ex[c] = expf(L[c] - mx); s += ex[c]; }
        float inv = 1.f / s;
        #pragma unroll
        for (int c = 0; c < 6; ++c) L[c] = ex[c] * inv;
    }
    __syncthreads();

    const int p = tid & 127;          // fixed position per thread
    float al[6];
    #pragma unroll
    for (int c = 0; c < 6; ++c) al[c] = sLog[p * 17 + c];
    const float* encP = enc + b * ENC_BSTRIDE + posBase + p;
    float*       outP = out + b * ENC_BSTRIDE + posBase + p;
    for (int ch = (tid >> 7); ch < 512; ch += 2) {
        float aexp = 0.f;
        #pragma unroll
        for (int c = 0; c < 6; ++c) aexp = fmaf(sWe[ch * 6 + c], al[c], aexp);
        float e = encP[ch * 1024];
        float g = 1.f / (1.f + expf(-e * aexp));
        outP[ch * 1024] = e + g;
    }
}

// ---------------------------------------------------------------------------
extern "C" void kernel_launch(void* const* d_in, const int* in_sizes, int n_in,
                              void* d_out, int out_size, void* d_ws, size_t ws_size,
                              hipStream_t stream) {
    const float* enc     = (const float*)d_in[0];   // (4,512,32,32)
    const int*   bboxes  = (const int*)  d_in[1];   // (4,5,4)
    const float* Wproc   = (const float*)d_in[2];   // (512,6,3,3)
    const float* bproc   = (const float*)d_in[3];   // (512,)
    const float* Walpha  = (const float*)d_in[4];   // (6,512,1,1)
    const float* Wexpand = (const float*)d_in[5];   // (512,6,1,1)

    float* out    = (float*)d_out;                  // sag_out  [0, 2097152)
    float* onehot = out + SAG_ELEMS;                // one_hot  [2097152, 8388608)
    float* Xk     = out + SAG_ELEMS + OH_ELEMS;     // X_dec_k  [8388608, 10485760)

    k_rasterize<<<OH_ELEMS / 4 / 256, 256, 0, stream>>>(bboxes, onehot);
    k_xdeck<<<4 * 32, 256, 0, stream>>>(bboxes, Wproc, bproc, Xk);
    k_sag<<<32, 256, 0, stream>>>(enc, Walpha, Wexpand, Xk, out);
}